// GNN_Model_6906307412125
// MI455X (gfx1250) — compile-verified
//
#include <hip/hip_runtime.h>
#include <math.h>

typedef __attribute__((ext_vector_type(2))) float v2f;
typedef __attribute__((ext_vector_type(4))) float v4f;
typedef __attribute__((ext_vector_type(8))) float v8f;

#define K_DIM 128

// ---------------------------------------------------------------------------
// C[M,HOUT] = act(A[M,128] @ W[HOUT,128]^T + bias), fp32 WMMA 16x16x4.
// One wave per 16-row strip, 8 waves (128 rows) per workgroup.
// W staged in LDS with XOR swizzle -> conflict-free ds_load_b64 B fragments.
// ---------------------------------------------------------------------------
template <int HOUT, bool RELU>
__global__ __launch_bounds__(256) void gemm_wmma_kernel(
    const float* __restrict__ A, const float* __restrict__ W,
    const float* __restrict__ bias, float* __restrict__ C, int M) {
  constexpr int NT = HOUT / 16;
  __shared__ float wlds[HOUT * K_DIM];  // 64KB (HOUT=128) / 32KB (HOUT=64)

  // Cooperative swizzled load of W (row-major [HOUT][128]) into LDS.
  for (int i = threadIdx.x; i < HOUT * K_DIM / 4; i += blockDim.x) {
    int flat = i * 4;
    int n = flat >> 7;
    int k = flat & 127;
    v4f v = *(const v4f*)(W + flat);
    *(v4f*)&wlds[n * K_DIM + (k ^ ((n & 15) << 2))] = v;
  }
  __syncthreads();

  const int lane = threadIdx.x & 31;
  const int wave = threadIdx.x >> 5;
  const int ln = lane & 15;   // N / M position within half-wave
  const int lh = lane >> 4;   // half-wave select
  const int m0 = (blockIdx.x * 8 + wave) * 16;
  if (m0 >= M) return;  // wave-uniform exit; active waves keep EXEC all-ones

  v8f acc[NT];
#pragma unroll
  for (int t = 0; t < NT; ++t) acc[t] = {};

  // A fragment row for this lane (clamp: clamped rows only feed OOB C rows).
  int ar = m0 + ln;
  if (ar >= M) ar = M - 1;
  const float* arow = A + (size_t)ar * K_DIM + 2 * lh;
  const int swz = ln << 2;

#pragma unroll 4
  for (int kc = 0; kc < K_DIM; kc += 4) {
    v2f a = *(const v2f*)(arow + kc);  // A[m][kc+2*lh .. +1]
#pragma unroll
    for (int nt = 0; nt < NT; ++nt) {
      const int n = nt * 16 + ln;
      // B[k][n] = W[n][k]; swizzled LDS read, 8B aligned, bank-conflict-free.
      v2f b = *(const v2f*)&wlds[n * K_DIM + ((kc + 2 * lh) ^ swz)];
      acc[nt] = __builtin_amdgcn_wmma_f32_16x16x4_f32(
          false, a, false, b, (short)0, acc[nt], false, false);
    }
  }

  // Epilogue: C/D layout -> lane holds col n, rows m0 + r + 8*lh, r=0..7.
#pragma unroll
  for (int nt = 0; nt < NT; ++nt) {
    const int n = nt * 16 + ln;
    const float bv = bias[n];
#pragma unroll
    for (int r = 0; r < 8; ++r) {
      const int rowi = m0 + r + 8 * lh;
      if (rowi < M) {
        float v = acc[nt][r] + bv;
        if (RELU) v = fmaxf(v, 0.0f);
        C[(size_t)rowi * HOUT + n] = v;
      }
    }
  }
}

// ---------------------------------------------------------------------------
// Elementwise / edge kernels
// ---------------------------------------------------------------------------
__global__ void fill_ones_kernel(float* __restrict__ p, int n) {
  int i = blockIdx.x * blockDim.x + threadIdx.x;
  if (i < n) p[i] = 1.0f;
}

__global__ void count_edges_kernel(const int* __restrict__ row,
                                   float* __restrict__ cnt, int E) {
  for (int i = blockIdx.x * blockDim.x + threadIdx.x; i < E;
       i += gridDim.x * blockDim.x)
    atomicAdd(&cnt[row[i]], 1.0f);  // exact integer-valued fp32 adds
}

__global__ void recip_kernel(float* __restrict__ p, int n) {
  int i = blockIdx.x * blockDim.x + threadIdx.x;
  if (i < n) p[i] = 1.0f / p[i];
}

template <bool SCALE>
__global__ void copy_scale_kernel(const float* __restrict__ in,
                                  const float* __restrict__ inv,
                                  float* __restrict__ out, int n) {
  for (int i = blockIdx.x * blockDim.x + threadIdx.x; i < n;
       i += gridDim.x * blockDim.x)
    out[i] = SCALE ? in[i] * inv[i >> 7] : in[i];
}

__global__ void relu_kernel(float* __restrict__ p, int n) {
  for (int i = blockIdx.x * blockDim.x + threadIdx.x; i < n;
       i += gridDim.x * blockDim.x)
    p[i] = fmaxf(p[i], 0.0f);
}

// One thread per feature, one block (128 threads) per edge iteration:
// coalesced 512B gather of h[src], 512B of global_atomic_add_f32 into out[dst].
// Feature buffers (25.6MB) are L2-resident on MI455X (192MB L2).
template <bool SCALE>
__global__ __launch_bounds__(128) void scatter_add_kernel(
    const float* __restrict__ h, const int* __restrict__ src,
    const int* __restrict__ dst, const float* __restrict__ inv,
    float* __restrict__ out, int E) {
  const int f = threadIdx.x;
  for (int e = blockIdx.x; e < E; e += gridDim.x) {
    const int s = src[e];
    const int d = dst[e];
    float v = h[(size_t)s * 128 + f];
    if (SCALE) v *= inv[s];
    atomicAdd(&out[(size_t)d * 128 + f], v);
  }
}

// In-place log_softmax over rows of 64; one wave per row, 2 cols/lane,
// wave32 butterfly reductions.
__global__ __launch_bounds__(256) void log_softmax64_kernel(
    float* __restrict__ p, int M) {
  const int rowi = blockIdx.x * (blockDim.x >> 5) + (threadIdx.x >> 5);
  const int lane = threadIdx.x & 31;
  if (rowi >= M) return;
  float* r = p + (size_t)rowi * 64;
  float a = r[lane];
  float b = r[lane + 32];
  float m = fmaxf(a, b);
  for (int off = 16; off; off >>= 1) m = fmaxf(m, __shfl_xor(m, off));
  float s = expf(a - m) + expf(b - m);
  for (int off = 16; off; off >>= 1) s += __shfl_xor(s, off);
  const float ls = logf(s);
  r[lane] = a - m - ls;
  r[lane + 32] = b - m - ls;
}

// ---------------------------------------------------------------------------
// Host-side pipeline.
// NOTE: the attention softmax collapses exactly to uniform 1/outdeg weights
// (alpha depends only on `row`, softmax is segmented by `row`), so fl_att /
// segment_max / exp / segment_sum are algebraically eliminated.
// ---------------------------------------------------------------------------
extern "C" void kernel_launch(void* const* d_in, const int* in_sizes, int n_in,
                              void* d_out, int out_size, void* d_ws,
                              size_t ws_size, hipStream_t stream) {
  (void)n_in; (void)out_size; (void)ws_size;

  const float* x     = (const float*)d_in[0];
  const int*   ei    = (const int*)d_in[1];
  const float* fl_W  = (const float*)d_in[2];
  const float* fl_b  = (const float*)d_in[3];
  /* d_in[4] = fl_att : mathematically eliminated (uniform softmax)        */
  const float* g1_W1 = (const float*)d_in[5];
  const float* g1_b1 = (const float*)d_in[6];
  const float* g1_W2 = (const float*)d_in[7];
  const float* g1_b2 = (const float*)d_in[8];
  const float* g2_W1 = (const float*)d_in[9];
  const float* g2_b1 = (const float*)d_in[10];
  const float* g2_W2 = (const float*)d_in[11];
  const float* g2_b2 = (const float*)d_in[12];
  const float* out_W = (const float*)d_in[13];
  const float* out_b = (const float*)d_in[14];
  float* out = (float*)d_out;

  const int N = in_sizes[0] / 128;  // 50000
  const int E = in_sizes[1] / 2;    // 600000
  const int* row = ei;       // edge_index[0] (softmax group / GIN source)
  const int* col = ei + E;   // edge_index[1] (aggregation target)

  float* bufA = (float*)d_ws;                 // N*128
  float* bufB = bufA + (size_t)N * 128;       // N*128
  float* inv  = bufB + (size_t)N * 128;       // N

  const int NF = N * 128;
  const int gemmGrid = (N + 127) / 128;
  const int elemGrid = (NF + 255) / 256;
  const int nodeGrid = (N + 255) / 256;

  // --- FastLocalConv ---
  gemm_wmma_kernel<128, false><<<gemmGrid, 256, 0, stream>>>(x, fl_W, fl_b, bufA, N);
  fill_ones_kernel<<<nodeGrid, 256, 0, stream>>>(inv, N);          // self loop
  count_edges_kernel<<<2048, 256, 0, stream>>>(row, inv, E);       // + outdeg
  recip_kernel<<<nodeGrid, 256, 0, stream>>>(inv, N);              // 1/cnt
  copy_scale_kernel<true><<<elemGrid, 256, 0, stream>>>(bufA, inv, bufB, NF);  // self-loop term
  scatter_add_kernel<true><<<4096, 128, 0, stream>>>(bufA, row, col, inv, bufB, E);
  relu_kernel<<<elemGrid, 256, 0, stream>>>(bufB, NF);             // h1 in bufB

  // --- GIN layer 1 ---
  copy_scale_kernel<false><<<elemGrid, 256, 0, stream>>>(bufB, nullptr, bufA, NF);
  scatter_add_kernel<false><<<4096, 128, 0, stream>>>(bufB, row, col, nullptr, bufA, E);
  gemm_wmma_kernel<128, true><<<gemmGrid, 256, 0, stream>>>(bufA, g1_W1, g1_b1, bufB, N);
  gemm_wmma_kernel<128, true><<<gemmGrid, 256, 0, stream>>>(bufB, g1_W2, g1_b2, bufA, N);
  // h2 in bufA

  // --- GIN layer 2 ---
  copy_scale_kernel<false><<<elemGrid, 256, 0, stream>>>(bufA, nullptr, bufB, NF);
  scatter_add_kernel<false><<<4096, 128, 0, stream>>>(bufA, row, col, nullptr, bufB, E);
  gemm_wmma_kernel<128, true><<<gemmGrid, 256, 0, stream>>>(bufB, g2_W1, g2_b1, bufA, N);
  gemm_wmma_kernel<128, true><<<gemmGrid, 256, 0, stream>>>(bufA, g2_W2, g2_b2, bufB, N);
  // h3 in bufB

  // --- Output head + log_softmax ---
  gemm_wmma_kernel<64, false><<<gemmGrid, 256, 0, stream>>>(bufB, out_W, out_b, out, N);
  log_softmax64_kernel<<<(N + 7) / 8, 256, 0, stream>>>(out, N);
}